// KVCache_43404939493855
// MI455X (gfx1250) — compile-verified
//
#include <hip/hip_runtime.h>
#include <stdint.h>

// Reference geometry:
//   pos_ids : (16,) int32
//   k, v    : (1, 8, 16, 128)   f32
//   caches  : (1, 8, 8192, 128) f32
//   out     : [kout | vout] flat = 2 * 8*8192*128 f32
#define N_KV_HEADS 8
#define MAX_CTX    8192
#define HEAD_DIM   128
#define N_NEW      16
#define CACHE_ELEMS (N_KV_HEADS * MAX_CTX * HEAD_DIM)   // 8,388,608 floats = 32 MiB
#define CHUNK_F32   8192                                // 32 KiB TDM tile
#define CHUNKS_PER_TENSOR (CACHE_ELEMS / CHUNK_F32)     // 1024

// ---------------------------------------------------------------------------
// Bulk clone via the Tensor Data Mover (gfx1250): one 32 KiB tile per wave,
// global -> LDS -> global, tracked with TENSORcnt. 2048 independent waves keep
// the whole 64 MiB clone in flight; SIMDs stay idle (DMA does the work).
// ---------------------------------------------------------------------------
__global__ void __launch_bounds__(32)
kv_clone_kernel(const float* __restrict__ kc, const float* __restrict__ vc,
                float* __restrict__ out) {
  __shared__ float stage[CHUNK_F32];                    // 32 KiB staging tile

  const uint32_t b = blockIdx.x;
  const uint32_t t = b / CHUNKS_PER_TENSOR;             // 0 = K cache, 1 = V cache
  const uint32_t c = b % CHUNKS_PER_TENSOR;
  const float* src = (t == 0 ? kc : vc) + (size_t)c * CHUNK_F32;
  float*       dst = out + (size_t)t * CACHE_ELEMS + (size_t)c * CHUNK_F32;

#if defined(__gfx1250__) && defined(__has_builtin) && \
    __has_builtin(__builtin_amdgcn_tensor_load_to_lds) && \
    __has_builtin(__builtin_amdgcn_tensor_store_from_lds) && \
    __has_builtin(__builtin_amdgcn_s_wait_tensorcnt)
  typedef unsigned int u32x4 __attribute__((ext_vector_type(4)));
  typedef int          i32x4 __attribute__((ext_vector_type(4)));
  typedef int          i32x8 __attribute__((ext_vector_type(8)));

  // Flat-aperture LDS addresses carry the wave-relative byte offset in [31:0].
  const uint32_t lds_off = (uint32_t)(uintptr_t)(void*)stage;

  // ---- D# group 1 (shared by load and store) --------------------------------
  // d0: workgroup_mask=0 (not in cluster), data_size=2 (4-byte elements)
  // tensor_dim0 = CHUNK_F32, tensor_dim1 = 1 (tile fully in-bounds)
  // tile_dim0   = CHUNK_F32, tile_dim1 = 1, tile_dim2 = 0 (2-D tensor)
  // tensor_dim0_stride = CHUNK_F32 (irrelevant: single row)
  i32x8 g1;
  g1[0] = (2 << 16);
  g1[1] = (int)((CHUNK_F32 & 0xFFFF) << 16);                    // tensor_dim0[15:0]
  g1[2] = (int)(((CHUNK_F32 >> 16) & 0xFFFF) | (1u << 16));     // dim0 hi | tensor_dim1 lo
  g1[3] = (int)((CHUNK_F32 & 0xFFFF) << 16);                    // tensor_dim1 hi=0 | tile_dim0
  g1[4] = 1;                                                    // tile_dim1=1, tile_dim2=0
  g1[5] = CHUNK_F32;                                            // tensor_dim0_stride lo
  g1[6] = 0;
  g1[7] = 0;

  // ---- D# group 0: count=1, lds_addr, 57-bit global addr, type=2 ------------
  const uint64_t ga_src = (uint64_t)(uintptr_t)src;
  const uint64_t ga_dst = (uint64_t)(uintptr_t)dst;
  u32x4 g0l, g0s;
  g0l.x = 1u;  g0l.y = lds_off;
  g0l.z = (uint32_t)ga_src;
  g0l.w = ((uint32_t)(ga_src >> 32) & 0x01FFFFFFu) | (2u << 30);
  g0s.x = 1u;  g0s.y = lds_off;
  g0s.z = (uint32_t)ga_dst;
  g0s.w = ((uint32_t)(ga_dst >> 32) & 0x01FFFFFFu) | (2u << 30);

  i32x4 z4; z4[0] = z4[1] = z4[2] = z4[3] = 0;                  // groups 2/3 unused (2-D)

#if __clang_major__ >= 23
  i32x8 z8; z8[0]=z8[1]=z8[2]=z8[3]=z8[4]=z8[5]=z8[6]=z8[7]=0;
  __builtin_amdgcn_tensor_load_to_lds(g0l, g1, z4, z4, z8, 0);
  __builtin_amdgcn_s_wait_tensorcnt(0);                         // tile resident in LDS
  __builtin_amdgcn_tensor_store_from_lds(g0s, g1, z4, z4, z8, 0);
#else
  __builtin_amdgcn_tensor_load_to_lds(g0l, g1, z4, z4, 0);
  __builtin_amdgcn_s_wait_tensorcnt(0);                         // tile resident in LDS
  __builtin_amdgcn_tensor_store_from_lds(g0s, g1, z4, z4, 0);
#endif
  __builtin_amdgcn_s_wait_tensorcnt(0);                         // drain before endpgm
#else
  // Fallback (toolchain without TDM builtins): plain wide copy of this chunk.
  (void)stage;
  const float4* s4 = (const float4*)src;
  float4*       d4 = (float4*)dst;
  for (int i = threadIdx.x; i < CHUNK_F32 / 4; i += 32) d4[i] = s4[i];
#endif
}

// ---------------------------------------------------------------------------
// Scatter the 16 new rows. One thread per (tensor, head, dim-float4); the j
// loop runs positions in ascending order so duplicate pos_ids are last-wins,
// matching clone-then-scatter semantics. 128 KiB total — negligible.
// ---------------------------------------------------------------------------
__global__ void kv_scatter_kernel(const int* __restrict__ pos,
                                  const float4* __restrict__ k,
                                  const float4* __restrict__ v,
                                  float4* __restrict__ out) {
  const int HD4 = HEAD_DIM / 4;                         // 32
  int tid = blockIdx.x * blockDim.x + threadIdx.x;      // 0 .. 2*8*32-1 = 511
  if (tid >= 2 * N_KV_HEADS * HD4) return;
  const int d4 = tid % HD4;
  const int h  = (tid / HD4) % N_KV_HEADS;
  const int t  = tid / (HD4 * N_KV_HEADS);

  const float4* src = (t == 0 ? k : v);
  float4* dstbase = out + (size_t)t * (CACHE_ELEMS / 4)
                        + (size_t)h * MAX_CTX * HD4;
  #pragma unroll
  for (int j = 0; j < N_NEW; ++j) {
    const int p = pos[j];
    dstbase[(size_t)p * HD4 + d4] = src[(h * N_NEW + j) * HD4 + d4];
  }
}

extern "C" void kernel_launch(void* const* d_in, const int* in_sizes, int n_in,
                              void* d_out, int out_size, void* d_ws, size_t ws_size,
                              hipStream_t stream) {
  const int*   pos = (const int*)d_in[0];
  const float* k   = (const float*)d_in[1];
  const float* v   = (const float*)d_in[2];
  const float* kc  = (const float*)d_in[3];
  const float* vc  = (const float*)d_in[4];
  float* out = (float*)d_out;

  // Bulk clone: 2048 single-wave blocks, one 32 KiB TDM tile each.
  kv_clone_kernel<<<2 * CHUNKS_PER_TENSOR, 32, 0, stream>>>(kc, vc, out);
  // Then overwrite the 16 target rows per head (stream-ordered after clone).
  kv_scatter_kernel<<<2, 256, 0, stream>>>(pos, (const float4*)k,
                                           (const float4*)v, (float4*)out);
}